// PAM_68195490725957
// MI455X (gfx1250) — compile-verified
//
#include <hip/hip_runtime.h>
#include <hip/hip_bf16.h>
#include <math.h>

// ---------------------------------------------------------------------------
// PAM (parallax attention) for B=2, C=64, H=W=256, WS=8.
// All GEMMs (grouped 3x3 conv implicit GEMM, window attention) run on
// v_wmma_f32_16x16x4_f32 (fp32, matches reference precision).
// wave32 operand layouts (CDNA5 ISA):
//   A 16x4 : lane L -> m=L&15, k-pair = (L>>4)*2, v2f = {A[m][k], A[m][k+1]}
//   B 4x16 : lane L -> n=L&15, same k-pair,       v2f = {B[k][n], B[k+1][n]}
//   C 16x16: VGPR v -> row m = v + 8*(L>>4), col n = L&15
// K-strided operands are K-PAIR-PACKED in LDS ([kp][n][2]) so each operand is
// one aligned ds_load_b64; inner-loop addresses are base-ptr + constant so the
// offsets fold into the ds_load immediate (no per-step VALU).
// ---------------------------------------------------------------------------

typedef float v2f __attribute__((ext_vector_type(2)));
typedef float v8f __attribute__((ext_vector_type(8)));

__device__ __forceinline__ v8f wmma4(v2f a, v2f b, v8f c) {
  // 8 args: (neg_a, A, neg_b, B, c_mod, C, reuse_a, reuse_b)
  return __builtin_amdgcn_wmma_f32_16x16x4_f32(false, a, false, b, (short)0, c,
                                               false, false);
}

__device__ __forceinline__ v2f lds2(const float* p) {
  return *(const v2f*)p;  // 8B-aligned by construction
}

#define HW 65536  // 256*256
#define C4 256    // 4*C
#define CC 64     // C
#define NB 2      // batch

// ---------------------------------------------------------------------------
// Kernel 0: one-time repack of grouped-conv weights into the k-pair-packed
// per-(group,tap) LDS image: wp[((g*9+tap)*32+kp)*128 + oc*2 + (ic&1)]
//   = w[((g*64+oc)*64 + ic)*9 + tap],  ic = 2*kp + parity.
// 147456 elements; writes coalesced. grid: 576 x 256.
// ---------------------------------------------------------------------------
__global__ __launch_bounds__(256) void wpack_kernel(const float* __restrict__ w,
                                                    float* __restrict__ wp) {
  int e = blockIdx.x * 256 + threadIdx.x;  // < 147456
  int oc2 = e & 127;
  int rest = e >> 7;
  int kp = rest & 31;
  int gt = rest >> 5;  // g*9 + tap
  int tap = gt % 9;
  int g = gt / 9;
  int oc = oc2 >> 1;
  int ic = kp * 2 + (oc2 & 1);
  wp[e] = w[(((g * 64 + oc) * 64 + ic) * 9) + tap];
}

// ---------------------------------------------------------------------------
// Kernel 1: BN batch stats -> fused scale/shift. grid 256 x 256 threads.
// ---------------------------------------------------------------------------
__global__ __launch_bounds__(256) void bn_stats_kernel(
    const float* __restrict__ cat, const float* __restrict__ gamma,
    const float* __restrict__ beta, float* __restrict__ scale,
    float* __restrict__ shift) {
  __shared__ float s_sum[256];
  __shared__ float s_sq[256];
  const int ch = blockIdx.x;
  const int t = threadIdx.x;
  float s = 0.f, q = 0.f;
  for (int b = 0; b < NB; ++b) {
    const float* p = cat + (((size_t)b * C4 + ch) << 16);
    for (int i = t; i < HW; i += 256) {
      float v = p[i];
      s += v;
      q += v * v;
    }
  }
  s_sum[t] = s;
  s_sq[t] = q;
  __syncthreads();
  for (int off = 128; off > 0; off >>= 1) {
    if (t < off) {
      s_sum[t] += s_sum[t + off];
      s_sq[t] += s_sq[t + off];
    }
    __syncthreads();
  }
  if (t == 0) {
    const float n = (float)(NB * HW);
    float mu = s_sum[0] / n;
    float var = s_sq[0] / n - mu * mu;
    float sc = gamma[ch] * rsqrtf(var + 1e-5f);
    scale[ch] = sc;
    shift[ch] = beta[ch] - mu * sc;
  }
}

// ---------------------------------------------------------------------------
// Kernel 2: grouped 3x3 conv (groups=4, 256->256), implicit GEMM via WMMA.
// Block = 128 threads (4 waves). Output: 64 oc (one group) x 32 px.
// Wave wv owns oc-tile wv*16 and BOTH 16-px tiles -> A fetched once/k-step.
// Weights come pre-packed (wpack_kernel), staged per tap as float4 copies.
// grid: (8 wtiles, 256 rows, B*4 groups)
// ---------------------------------------------------------------------------
template <bool IN_BN, bool LRELU, bool RESID>
__global__ __launch_bounds__(128) void conv3x3_kernel(
    const float* __restrict__ src, const float* __restrict__ cat,
    const float* __restrict__ wpk, const float* __restrict__ bias,
    const float* __restrict__ scale, const float* __restrict__ shift,
    float* __restrict__ dst) {
  // Xs: k-pair packed input tile [row r][kp][col][2], col c <-> w = w0-1+c
  __shared__ __attribute__((aligned(16))) float Xs[3 * 32 * 34 * 2];  // 26112B
  // Wt: k-pair packed weights for one tap [kp][oc][2]
  __shared__ __attribute__((aligned(16))) float Wt[32 * 64 * 2];      // 16384B

  const int t = threadIdx.x;
  const int w0 = blockIdx.x * 32;
  const int h0 = blockIdx.y;
  const int bb = blockIdx.z >> 2;
  const int g = blockIdx.z & 3;

  const float* wgrp = wpk + (size_t)g * 9 * 4096;  // this group's packed block
  // Warm L2 (global_prefetch_b8), contiguous 144KB.
  for (int e = t * 32; e < 9 * 4096; e += 128 * 32)
    __builtin_prefetch(&wgrp[e], 0, 1);

  // Stage input tile (zero padded; BN fused for conv1; padding stays 0,
  // matching conv-after-bn zero padding in the reference).
  for (int e = t; e < 3 * 64 * 34; e += 128) {
    int r = e / (64 * 34);
    int rm = e - r * (64 * 34);
    int ic = rm / 34;
    int cx = rm - ic * 34;
    int hh = h0 - 1 + r;
    int ww = w0 - 1 + cx;
    int ch = g * 64 + ic;
    float v = 0.f;
    if (hh >= 0 && hh < 256 && ww >= 0 && ww < 256) {
      v = src[(((size_t)bb * C4 + ch) << 16) + hh * 256 + ww];
      if (IN_BN) v = fmaf(v, scale[ch], shift[ch]);
    }
    Xs[((r * 32 + (ic >> 1)) * 34 + cx) * 2 + (ic & 1)] = v;
  }

  const int lane = t & 31;
  const int wv = t >> 5;  // 0..3 -> oc tile
  const int ma = lane & 15;
  const int half = lane >> 4;
  const int mbase = wv * 16;

  v8f acc0 = 0.0f;  // px tile 0..15
  v8f acc1 = 0.0f;  // px tile 16..31

  // Loop-invariant LDS base pointers (per-step offset is a constant).
  const float* aptr = &Wt[half * 128 + (mbase + ma) * 2];

  for (int tap = 0; tap < 9; ++tap) {
    const int dy = tap / 3;
    const int dx = tap - dy * 3;
    __syncthreads();  // prior tap done reading Wt (also guards Xs staging)
    {
      const float4* wsrc = (const float4*)(wgrp + tap * 4096);
      float4* wdst = (float4*)Wt;
      for (int e = t; e < 1024; e += 128) wdst[e] = wsrc[e];
    }
    __syncthreads();
    const float* bptr0 = &Xs[dy * 2176 + half * 68 + (ma + dx) * 2];
    const float* bptr1 = bptr0 + 32;  // +16 cols
#pragma unroll
    for (int k0 = 0; k0 < 64; k0 += 4) {
      v2f a = lds2(aptr + k0 * 64);
      v2f b0 = lds2(bptr0 + k0 * 34);
      acc0 = wmma4(a, b0, acc0);
      v2f b1 = lds2(bptr1 + k0 * 34);
      acc1 = wmma4(a, b1, acc1);
    }
  }

  // Epilogue: bias (+LeakyReLU) (+BN residual), store both px tiles.
#pragma unroll
  for (int v = 0; v < 8; ++v) {
    int ocl = mbase + v + 8 * half;  // local oc within group
    int oc = g * 64 + ocl;
    float bval = bias[oc];
    {
      int ww = w0 + ma;
      size_t o = (((size_t)bb * C4 + oc) << 16) + h0 * 256 + ww;
      float r = acc0[v] + bval;
      if (LRELU) r = r > 0.f ? r : 0.1f * r;
      if (RESID) r += fmaf(cat[o], scale[oc], shift[oc]);
      dst[o] = r;
    }
    {
      int ww = w0 + 16 + ma;
      size_t o = (((size_t)bb * C4 + oc) << 16) + h0 * 256 + ww;
      float r = acc1[v] + bval;
      if (LRELU) r = r > 0.f ? r : 0.1f * r;
      if (RESID) r += fmaf(cat[o], scale[oc], shift[oc]);
      dst[o] = r;
    }
  }
}

// ---------------------------------------------------------------------------
// Kernel 3: grouped 1x1 conv (groups=4, 256->64). grid: (1024 px tiles, B).
// ---------------------------------------------------------------------------
__global__ __launch_bounds__(256) void conv1x1_kernel(
    const float* __restrict__ src, const float* __restrict__ wgt,
    const float* __restrict__ bias, float* __restrict__ dst) {
  __shared__ float Ws[64][64];  // [oc][ic]
  const int t = threadIdx.x;
  for (int e = t; e < 4096; e += 256) Ws[e >> 6][e & 63] = wgt[e];
  __syncthreads();
  const int bb = blockIdx.y;
  const int px = blockIdx.x * 64 + (t & 63);
  const int go = t >> 6;  // group 0..3
  float acc[16];
#pragma unroll
  for (int j = 0; j < 16; ++j) acc[j] = bias[go * 16 + j];
  for (int ic = 0; ic < 64; ++ic) {
    float v = src[(((size_t)bb * C4 + go * 64 + ic) << 16) + px];
#pragma unroll
    for (int j = 0; j < 16; ++j) acc[j] = fmaf(Ws[go * 16 + j][ic], v, acc[j]);
  }
#pragma unroll
  for (int j = 0; j < 16; ++j)
    dst[(((size_t)bb * CC + go * 16 + j) << 16) + px] = acc[j];
}

// ---------------------------------------------------------------------------
// Kernel 4: per-window attention, both directions.
// Scrambled patchify-of-gather closed form (from the reference's flat
// reinterpretation of [b,h,w,c] as [b,c,hn,WS,wn,WS]):
//   hh = 4*q + (hi>>3); ww = 32*(hi&7) + 4*i + (wi>>3); cc = 8*(wi&7) + j
// grid: (wn=32, hn=32, B*2); z = bb + 2*dir. dir0 -> out_left, dir1 -> out_right.
// LDS: As [64][68] row-major (A window -> later Out), Bp k-pair packed
// (B window -> later X_sel), Ps [64][68] (scores -> probs, A of GEMM2).
// ---------------------------------------------------------------------------
__global__ __launch_bounds__(256) void pam_attn_kernel(
    const float* __restrict__ Qf, const float* __restrict__ Kf,
    const float* __restrict__ xl, const float* __restrict__ xr,
    const int* __restrict__ dl, const int* __restrict__ dr,
    float* __restrict__ outL, float* __restrict__ outR) {
  __shared__ __attribute__((aligned(16))) float As[64 * 68];      // 17408B
  __shared__ __attribute__((aligned(16))) float Ps[64 * 68];      // 17408B
  __shared__ __attribute__((aligned(16))) float Bp[32 * 64 * 2];  // 16384B

  const int t = threadIdx.x;
  const int wi = blockIdx.x;
  const int hi = blockIdx.y;
  const int bb = blockIdx.z & 1;
  const int dir = blockIdx.z >> 1;

  const float* Af = dir ? Kf : Qf;
  const float* Bf = dir ? Qf : Kf;
  const float* xga = dir ? xl : xr;  // gathered value tensor
  const float* xba = dir ? xr : xl;  // residual base
  const int* dsp = dir ? dr : dl;
  float* op = dir ? outR : outL;

  // A window (plain patchify): As[p][q] = Af[bb, q, hi*8+i, wi*8+j]
  for (int e = t; e < 4096; e += 256) {
    int q = e >> 6, p = e & 63;
    int i = p >> 3, j = p & 7;
    As[p * 68 + q] =
        Af[(((size_t)bb * CC + q) << 16) + (hi * 8 + i) * 256 + (wi * 8 + j)];
  }
  // B window (scrambled gather), stored k-pair packed: Bp[q>>1][p'][q&1]
  for (int e = t; e < 4096; e += 256) {
    int q = e >> 6, p = e & 63;
    int i = p >> 3, j = p & 7;
    int hh = 4 * q + (hi >> 3);
    int ww = ((hi & 7) << 5) + (i << 2) + (wi >> 3);
    int cc = ((wi & 7) << 3) + j;
    int d = dsp[((size_t)bb << 16) + hh * 256 + ww];
    int wsrc = dir ? (ww + d > 255 ? 255 : ww + d) : (ww - d < 0 ? 0 : ww - d);
    Bp[((q >> 1) * 64 + p) * 2 + (q & 1)] =
        Bf[(((size_t)bb * CC + cc) << 16) + hh * 256 + wsrc];
  }
  __syncthreads();
  // Center B rows (mean over the 64 patch entries per channel q).
  if (t < 64) {
    const int base = (t >> 1) * 128 + (t & 1);
    float s = 0.f;
    for (int p = 0; p < 64; ++p) s += Bp[base + 2 * p];
    float m = s * (1.0f / 64.0f);
    for (int p = 0; p < 64; ++p) Bp[base + 2 * p] -= m;
  }
  __syncthreads();

  const int lane = t & 31;
  const int wv = t >> 5;
  const int mtile = wv >> 1;
  const int n0 = (wv & 1) * 32;
  const int ma = lane & 15;
  const int half = lane >> 4;
  const int mbase = mtile * 16;

  // Loop-invariant LDS base pointers.
  const float* a1p = &As[(mbase + ma) * 68 + 2 * half];
  const float* a2p = &Ps[(mbase + ma) * 68 + 2 * half];
  const float* bbp = &Bp[half * 128 + (n0 + ma) * 2];

  // GEMM1: score[p][p'] = sum_q As[p][q] * B[q][p']  (each wave: 2 tiles)
  v8f acc0 = 0.0f, acc1 = 0.0f;
#pragma unroll
  for (int k0 = 0; k0 < 64; k0 += 4) {
    v2f a = lds2(a1p + k0);
    v2f b0 = lds2(bbp + k0 * 64);
    acc0 = wmma4(a, b0, acc0);
    v2f b1 = lds2(bbp + k0 * 64 + 32);
    acc1 = wmma4(a, b1, acc1);
  }
#pragma unroll
  for (int v = 0; v < 8; ++v) {
    Ps[(mbase + v + 8 * half) * 68 + n0 + ma] = acc0[v];
    Ps[(mbase + v + 8 * half) * 68 + n0 + 16 + ma] = acc1[v];
  }
  __syncthreads();

  // Softmax per row p; concurrently stage X_sel (k-pair packed into Bp, free
  // now) for GEMM2.
  if (t < 64) {
    float* row = &Ps[t * 68];
    float mx = -1e30f;
    for (int p = 0; p < 64; ++p) mx = fmaxf(mx, row[p]);
    float s = 0.f;
    for (int p = 0; p < 64; ++p) {
      float e = __expf(row[p] - mx);
      row[p] = e;
      s += e;
    }
    float inv = 1.0f / s;
    for (int p = 0; p < 64; ++p) row[p] *= inv;
  }
  for (int e = t; e < 4096; e += 256) {
    int c2 = e & 63, p = e >> 6;
    int i = p >> 3, j = p & 7;
    int hh = 4 * c2 + (hi >> 3);
    int ww = ((hi & 7) << 5) + (i << 2) + (wi >> 3);
    int cc = ((wi & 7) << 3) + j;
    int d = dsp[((size_t)bb << 16) + hh * 256 + ww];
    int wsrc = dir ? (ww + d > 255 ? 255 : ww + d) : (ww - d < 0 ? 0 : ww - d);
    Bp[((p >> 1) * 64 + c2) * 2 + (p & 1)] =
        xga[(((size_t)bb * CC + cc) << 16) + hh * 256 + wsrc];
  }
  __syncthreads();

  // GEMM2: Out[p][c2] = sum_p' Ps[p][p'] * Xsel[p'][c2]; Out -> As (free).
  v8f oa0 = 0.0f, oa1 = 0.0f;
#pragma unroll
  for (int k0 = 0; k0 < 64; k0 += 4) {
    v2f a = lds2(a2p + k0);
    v2f b0 = lds2(bbp + k0 * 64);
    oa0 = wmma4(a, b0, oa0);
    v2f b1 = lds2(bbp + k0 * 64 + 32);
    oa1 = wmma4(a, b1, oa1);
  }
#pragma unroll
  for (int v = 0; v < 8; ++v) {
    As[(mbase + v + 8 * half) * 68 + n0 + ma] = oa0[v];
    As[(mbase + v + 8 * half) * 68 + n0 + 16 + ma] = oa1[v];
  }
  __syncthreads();

  // Epilogue: out = x + Out * mask (coalesced 8-float rows per (c2,i)).
  for (int e = t; e < 4096; e += 256) {
    int c2 = e >> 6, p = e & 63;
    int i = p >> 3, j = p & 7;
    int hh = hi * 8 + i;
    int ww = wi * 8 + j;
    size_t o = (((size_t)bb * CC + c2) << 16) + hh * 256 + ww;
    int d = dsp[((size_t)bb << 16) + hh * 256 + ww];
    bool valid = dir ? (ww + d <= 255) : (ww - d >= 0);
    op[o] = xba[o] + (valid ? As[p * 68 + c2] : 0.0f);
  }
}

// ---------------------------------------------------------------------------
// Host orchestration.
// Workspace (floats): [scale/shift L+R: 1024] [y1: 2*256*65536]
// [resb: 2*256*65536] [Q: 2*64*65536] [K: 2*64*65536] [wp1: 147456]
// [wp2: 147456]   (~337 MB)
// ---------------------------------------------------------------------------
extern "C" void kernel_launch(void* const* d_in, const int* in_sizes, int n_in,
                              void* d_out, int out_size, void* d_ws,
                              size_t ws_size, hipStream_t stream) {
  (void)in_sizes; (void)n_in; (void)out_size; (void)ws_size;
  const float* x_l = (const float*)d_in[0];
  const float* x_r = (const float*)d_in[1];
  const float* cat_l = (const float*)d_in[2];
  const float* cat_r = (const float*)d_in[3];
  const int* d_l = (const int*)d_in[4];
  const int* d_r = (const int*)d_in[5];
  const float* gamma = (const float*)d_in[6];
  const float* beta = (const float*)d_in[7];
  const float* w1 = (const float*)d_in[8];
  const float* b1 = (const float*)d_in[9];
  const float* w2 = (const float*)d_in[10];
  const float* b2 = (const float*)d_in[11];
  const float* bqw = (const float*)d_in[12];
  const float* bqb = (const float*)d_in[13];
  const float* bsw = (const float*)d_in[14];
  const float* bsb = (const float*)d_in[15];

  float* ws = (float*)d_ws;
  float* scaleL = ws;
  float* shiftL = ws + 256;
  float* scaleR = ws + 512;
  float* shiftR = ws + 768;
  float* y1 = ws + 1024;
  const size_t featN = (size_t)NB * C4 * HW;  // 33,554,432
  float* rb = y1 + featN;
  float* Qb = rb + featN;
  float* Kb = Qb + (size_t)NB * CC * HW;
  float* wp1 = Kb + (size_t)NB * CC * HW;
  float* wp2 = wp1 + 147456;

  float* outL = (float*)d_out;
  float* outR = outL + (size_t)NB * CC * HW;

  // One-time weight repack + BN stats.
  wpack_kernel<<<576, 256, 0, stream>>>(w1, wp1);
  wpack_kernel<<<576, 256, 0, stream>>>(w2, wp2);
  bn_stats_kernel<<<256, 256, 0, stream>>>(cat_l, gamma, beta, scaleL, shiftL);
  bn_stats_kernel<<<256, 256, 0, stream>>>(cat_r, gamma, beta, scaleR, shiftR);

  dim3 cg(8, 256, NB * 4);
  // Left branch: Q = bq(resb(bn(cat_l)))
  conv3x3_kernel<true, true, false><<<cg, 128, 0, stream>>>(
      cat_l, nullptr, wp1, b1, scaleL, shiftL, y1);
  conv3x3_kernel<false, false, true><<<cg, 128, 0, stream>>>(
      y1, cat_l, wp2, b2, scaleL, shiftL, rb);
  conv1x1_kernel<<<dim3(1024, NB), 256, 0, stream>>>(rb, bqw, bqb, Qb);
  // Right branch: K = bs(resb(bn(cat_r)))
  conv3x3_kernel<true, true, false><<<cg, 128, 0, stream>>>(
      cat_r, nullptr, wp1, b1, scaleR, shiftR, y1);
  conv3x3_kernel<false, false, true><<<cg, 128, 0, stream>>>(
      y1, cat_r, w2 ? wp2 : wp2, b2, scaleR, shiftR, rb);
  conv1x1_kernel<<<dim3(1024, NB), 256, 0, stream>>>(rb, bsw, bsb, Kb);

  // Window attention, both directions.
  pam_attn_kernel<<<dim3(32, 32, NB * 2), 256, 0, stream>>>(
      Qb, Kb, x_l, x_r, d_l, d_r, outL, outR);
}